// ContinuousDepthGeneModule_32083405701714
// MI455X (gfx1250) — compile-verified
//
#include <hip/hip_runtime.h>
#include <cstdint>

#define HIDDEN 128

typedef float v2f __attribute__((ext_vector_type(2)));
typedef float v8f __attribute__((ext_vector_type(8)));

// ---------------------------------------------------------------------------
// Scalars: epigenetic scale + clamped depth
// ---------------------------------------------------------------------------
__global__ void scalars_kernel(const float* __restrict__ meth,
                               const float* __restrict__ hist,
                               const float* __restrict__ logd,
                               float* __restrict__ sc) {
    __shared__ float red[HIDDEN];
    const int t = threadIdx.x;
    red[t] = 1.f / (1.f + expf(-meth[t]));
    __syncthreads();
    for (int s = HIDDEN / 2; s > 0; s >>= 1) {
        if (t < s) red[t] += red[t + s];
        __syncthreads();
    }
    if (t == 0) {
        float meth_sil = red[0] * (1.f / (float)HIDDEN);
        float h0 = 1.f / (1.f + expf(-hist[0]));
        float h1 = 1.f / (1.f + expf(-hist[1]));
        float h2 = 1.f / (1.f + expf(-hist[2]));
        float h3 = 1.f / (1.f + expf(-hist[3]));
        float act = (h0 + h2) * 0.5f, rep = (h1 + h3) * 0.5f;
        float chrom = fminf(fmaxf(act - rep + 0.5f, 0.f), 1.f);
        sc[0] = chrom * (1.f - meth_sil);                 // feature scale
        sc[1] = fminf(fmaxf(expf(logd[0]), 0.1f), 3.0f);  // dt = depth
    }
}

// ---------------------------------------------------------------------------
// Degree / normalization for symmetric GCN
// ---------------------------------------------------------------------------
__global__ void degree_kernel(const int* __restrict__ dst, float* __restrict__ deg, int ne) {
    int e = blockIdx.x * 256 + threadIdx.x;
    if (e >= ne) return;
    atomicAdd(&deg[dst[e]], 1.f);
}

__global__ void dinv_kernel(float* __restrict__ deg, int n) {
    int i = blockIdx.x * 256 + threadIdx.x;
    if (i >= n) return;
    float d = deg[i];
    deg[i] = (d > 0.f) ? rsqrtf(fmaxf(d, 1.f)) : 0.f;
}

__global__ void enorm_kernel(const int* __restrict__ src, const int* __restrict__ dst,
                             const float* __restrict__ dinv, float* __restrict__ enorm, int ne) {
    int e = blockIdx.x * 256 + threadIdx.x;
    if (e >= ne) return;
    enorm[e] = dinv[src[e]] * dinv[dst[e]];
}

// ---------------------------------------------------------------------------
// Wave-level f32 WMMA GEMM: out[N x 128] = A[N x K] @ W[K x 128]  (+epilogue)
//   MODE 0: plain store            MODE 1: bias+LN+ReLU+scale (input proj)
//   MODE 2: gate blend (K=256, A=cur | A2=h_new, out=cur in-place)
//   MODE 3: bias+LN + atomic pool  (output proj)
// One wave owns a 16-row x 128-col strip. W staged through LDS in 32-row
// chunks with row-pair interleaving: lds[(k>>1)*256 + n*2 + (k&1)] so a
// B fragment (W[k][n], W[k+1][n]) is one aligned ds_load_b64.
// ---------------------------------------------------------------------------
template <int K, int MODE>
__global__ __launch_bounds__(256) void gemm128(
    const float* __restrict__ A, const float* __restrict__ A2,
    const float* __restrict__ W, const float* __restrict__ bias,
    const float* __restrict__ lng, const float* __restrict__ lnb,
    const float* __restrict__ scalars, const int* __restrict__ batchv,
    float* __restrict__ pool, float* __restrict__ out, int nrows) {
    __shared__ float ldsW[32 * HIDDEN];
    const int tid = threadIdx.x;
    const int lane = tid & 31;
    const int wid = tid >> 5;
    const int m = lane & 15;          // row (A) / col (B,C) within tile
    const int kb = (lane >> 4) << 1;  // 0 or 2 : K sub-offset per half-wave
    const int row0 = blockIdx.x * 128 + wid * 16;
    const int row = row0 + m;
    const bool rv = row < nrows;

    v8f acc[8] = {};

    for (int k0 = 0; k0 < K; k0 += 32) {
        // stage 32x128 chunk of W, float4 global loads, interleaved LDS stores
#pragma unroll
        for (int it = 0; it < 4; ++it) {
            const int flat = tid + it * 256;        // float4 slot id (0..1023)
            const int k = flat >> 5;                // chunk row 0..31
            const int c4 = flat & 31;               // float4 col 0..31
            const float4 v = *(const float4*)(W + (size_t)(k0 + k) * HIDDEN + c4 * 4);
            const int base = (k >> 1) * 256 + (c4 * 4) * 2 + (k & 1);
            ldsW[base + 0] = v.x;
            ldsW[base + 2] = v.y;
            ldsW[base + 4] = v.z;
            ldsW[base + 6] = v.w;
        }
        __syncthreads();
#pragma unroll
        for (int kk = 0; kk < 32; kk += 4) {
            const int kcol = k0 + kk + kb;
            v2f a;
            if (rv) {
                const float* ap;
                if (MODE == 2)
                    ap = (kcol < HIDDEN) ? (A + (size_t)row * HIDDEN + kcol)
                                         : (A2 + (size_t)row * HIDDEN + (kcol - HIDDEN));
                else
                    ap = A + (size_t)row * K + kcol;
                a = *(const v2f*)ap;
            } else {
                a.x = 0.f;
                a.y = 0.f;
            }
            const int ib = ((kk + kb) >> 1) * 256;  // interleaved pair-row base
            v2f bf[8];
#pragma unroll
            for (int t = 0; t < 8; ++t)
                bf[t] = *(const v2f*)&ldsW[ib + (t * 16 + m) * 2];
#pragma unroll
            for (int t = 0; t < 8; ++t)
                acc[t] = __builtin_amdgcn_wmma_f32_16x16x4_f32(
                    false, a, false, bf[t], (short)0, acc[t], false, false);
        }
        __syncthreads();
    }

    // C/D layout: half h = lane>>4; VGPR r holds row m = r + 8*h; col n = t*16 + (lane&15)
    const int hb = (lane >> 4) << 3;
    if (MODE == 0) {
#pragma unroll
        for (int r = 0; r < 8; ++r) {
            const int orow = row0 + r + hb;
            if (orow < nrows) {
#pragma unroll
                for (int t = 0; t < 8; ++t)
                    out[(size_t)orow * HIDDEN + t * 16 + m] = acc[t][r];
            }
        }
    } else if (MODE == 2) {
#pragma unroll
        for (int r = 0; r < 8; ++r) {
            const int orow = row0 + r + hb;
            if (orow < nrows) {
#pragma unroll
                for (int t = 0; t < 8; ++t) {
                    const int n = t * 16 + m;
                    const size_t o = (size_t)orow * HIDDEN + n;
                    float g = 1.f / (1.f + expf(-(acc[t][r] + bias[n])));
                    float hn = A2[o];
                    float cu = A[o];
                    out[o] = g * hn + (1.f - g) * cu;
                }
            }
        }
    } else {  // MODE 1 / MODE 3 : bias + LayerNorm epilogue
        const float scl = (MODE == 1) ? scalars[0] : 1.f;
#pragma unroll
        for (int r = 0; r < 8; ++r) {
            const int orow = row0 + r + hb;
            float vals[8];
            float s = 0.f;
#pragma unroll
            for (int t = 0; t < 8; ++t) {
                float v = acc[t][r] + bias[t * 16 + m];
                vals[t] = v;
                s += v;
            }
            s += __shfl_xor(s, 1, 32); s += __shfl_xor(s, 2, 32);
            s += __shfl_xor(s, 4, 32); s += __shfl_xor(s, 8, 32);
            const float mean = s * (1.f / 128.f);
            float s2 = 0.f;
#pragma unroll
            for (int t = 0; t < 8; ++t) {
                float d = vals[t] - mean;
                s2 += d * d;
            }
            s2 += __shfl_xor(s2, 1, 32); s2 += __shfl_xor(s2, 2, 32);
            s2 += __shfl_xor(s2, 4, 32); s2 += __shfl_xor(s2, 8, 32);
            const float inv = rsqrtf(s2 * (1.f / 128.f) + 1e-5f);
            if (orow < nrows) {
                const int gidx = (MODE == 3) ? batchv[orow] : 0;
#pragma unroll
                for (int t = 0; t < 8; ++t) {
                    const int n = t * 16 + m;
                    float v = (vals[t] - mean) * inv * lng[n] + lnb[n];
                    if (MODE == 1) {
                        v = fmaxf(v, 0.f) * scl;
                        out[(size_t)orow * HIDDEN + n] = v;
                    } else {
                        atomicAdd(&pool[gidx * HIDDEN + n], v);
                    }
                }
            }
        }
    }
}

// ---------------------------------------------------------------------------
// Edge aggregation: one wave per edge (edge id scalarized via readfirstlane),
// float4 per lane (coalesced 512B row read), L2 atomics on dst rows.
// ---------------------------------------------------------------------------
__global__ void aggregate_kernel(const float* __restrict__ xw, const int* __restrict__ src,
                                 const int* __restrict__ dst, const float* __restrict__ enorm,
                                 float* __restrict__ agg, int ne) {
    const int gid = blockIdx.x * 256 + threadIdx.x;
    const int e = __builtin_amdgcn_readfirstlane(gid >> 5);  // wave-uniform edge id
    const int c = threadIdx.x & 31;
    if (e >= ne) return;
    const int s = src[e], d = dst[e];
    const float w = enorm[e];
    const float4 v = *((const float4*)(xw + (size_t)s * HIDDEN) + c);
    float* ap = agg + (size_t)d * HIDDEN + c * 4;
    atomicAdd(ap + 0, v.x * w);
    atomicAdd(ap + 1, v.y * w);
    atomicAdd(ap + 2, v.z * w);
    atomicAdd(ap + 3, v.w * w);
}

// ---------------------------------------------------------------------------
// Per-row bias + LayerNorm (one wave per row, float4 per lane)
// ---------------------------------------------------------------------------
__global__ void bias_ln_kernel(const float* __restrict__ in, const float* __restrict__ bias,
                               const float* __restrict__ g, const float* __restrict__ b,
                               float* __restrict__ out, int nrows) {
    int gw = (blockIdx.x * 256 + threadIdx.x) >> 5;
    int lane = threadIdx.x & 31;
    if (gw >= nrows) return;
    size_t base = (size_t)gw * HIDDEN + lane * 4;
    float4 v = *(const float4*)(in + base);
    float4 bb = *(const float4*)(bias + lane * 4);
    float x0 = v.x + bb.x, x1 = v.y + bb.y, x2 = v.z + bb.z, x3 = v.w + bb.w;
    float s = x0 + x1 + x2 + x3;
    s += __shfl_xor(s, 1, 32); s += __shfl_xor(s, 2, 32);
    s += __shfl_xor(s, 4, 32); s += __shfl_xor(s, 8, 32); s += __shfl_xor(s, 16, 32);
    float mean = s * (1.f / 128.f);
    float d0 = x0 - mean, d1 = x1 - mean, d2 = x2 - mean, d3 = x3 - mean;
    float s2 = d0 * d0 + d1 * d1 + d2 * d2 + d3 * d3;
    s2 += __shfl_xor(s2, 1, 32); s2 += __shfl_xor(s2, 2, 32);
    s2 += __shfl_xor(s2, 4, 32); s2 += __shfl_xor(s2, 8, 32); s2 += __shfl_xor(s2, 16, 32);
    float inv = rsqrtf(s2 * (1.f / 128.f) + 1e-5f);
    float4 gg = *(const float4*)(g + lane * 4);
    float4 be = *(const float4*)(b + lane * 4);
    float4 o;
    o.x = d0 * inv * gg.x + be.x;
    o.y = d1 * inv * gg.y + be.y;
    o.z = d2 * inv * gg.z + be.z;
    o.w = d3 * inv * gg.w + be.w;
    *(float4*)(out + base) = o;
}

// ---------------------------------------------------------------------------
// Fused tanh + residual + RK4 stage update:
//   k      = tanh(cur) + rw * yin
//   y_out  = h0 + c*dt*k
//   acc   += w*(dt/6)*k
// ---------------------------------------------------------------------------
__global__ void tanh_rk4_kernel(const float* __restrict__ cur, const float* __restrict__ yin,
                                const float* __restrict__ h0, float* __restrict__ yout,
                                float* __restrict__ acc, const float* __restrict__ rw,
                                const float* __restrict__ sc, float c, float w, int n) {
    int i = blockIdx.x * 256 + threadIdx.x;
    if (i >= n) return;
    float dt = sc[1];
    float k = tanhf(cur[i]) + rw[0] * yin[i];
    yout[i] = h0[i] + c * dt * k;
    acc[i] += w * (dt * (1.f / 6.f)) * k;
}

__global__ void count_kernel(const int* __restrict__ batchv, float* __restrict__ cnt, int n) {
    int i = blockIdx.x * 256 + threadIdx.x;
    if (i >= n) return;
    atomicAdd(&cnt[batchv[i]], 1.f);
}

__global__ void finalize_kernel(const float* __restrict__ pool, const float* __restrict__ cnt,
                                float* __restrict__ outp, int n) {
    int i = blockIdx.x * 256 + threadIdx.x;
    if (i >= n) return;
    outp[i] = pool[i] / fmaxf(cnt[i >> 7], 1.f);
}

// ---------------------------------------------------------------------------
// Host orchestration
// ---------------------------------------------------------------------------
extern "C" void kernel_launch(void* const* d_in, const int* in_sizes, int n_in,
                              void* d_out, int out_size, void* d_ws, size_t ws_size,
                              hipStream_t stream) {
    (void)n_in; (void)ws_size;
    const float* x      = (const float*)d_in[0];
    const int*   eidx   = (const int*)d_in[1];
    const int*   batchv = (const int*)d_in[2];
    const float* in_w   = (const float*)d_in[3];
    const float* in_b   = (const float*)d_in[4];
    const float* in_lg  = (const float*)d_in[5];
    const float* in_lb  = (const float*)d_in[6];
    const float* meth   = (const float*)d_in[7];
    const float* hist   = (const float*)d_in[8];
    const float* logd   = (const float*)d_in[9];
    const float* gcn_w  = (const float*)d_in[10];
    const float* gcn_b  = (const float*)d_in[11];
    const float* ln_g   = (const float*)d_in[12];
    const float* ln_b   = (const float*)d_in[13];
    const float* gate_w = (const float*)d_in[14];
    const float* gate_b = (const float*)d_in[15];
    const float* rw     = (const float*)d_in[16];
    const float* out_w  = (const float*)d_in[17];
    const float* out_b  = (const float*)d_in[18];
    const float* o_lg   = (const float*)d_in[19];
    const float* o_lb   = (const float*)d_in[20];

    const int N = in_sizes[0] / 64;
    const int E = in_sizes[1] / 2;
    const int G = out_size / HIDDEN;
    const int NH = N * HIDDEN;
    const int* src = eidx;
    const int* dst = eidx + E;

    // workspace layout (floats)
    float* SC    = (float*)d_ws;       // 16 scalars: [0]=scale, [1]=dt
    float* DEG   = SC + 16;            // N (deg -> dinv in place)
    float* ENORM = DEG + N;            // E
    float* H0    = ENORM + E;          // N*128
    float* Y     = H0 + (size_t)NH;
    float* CUR   = Y + (size_t)NH;
    float* XW    = CUR + (size_t)NH;
    float* AGG   = XW + (size_t)NH;
    float* ACC   = AGG + (size_t)NH;
    float* POOL  = ACC + (size_t)NH;           // G*128
    float* CNT   = POOL + (size_t)G * HIDDEN;  // G

    const unsigned gGemm = (unsigned)((N + 127) / 128);
    const unsigned gLN   = (unsigned)((N + 7) / 8);
    const unsigned gEl   = (unsigned)((NH + 255) / 256);
    const unsigned gE    = (unsigned)((E + 255) / 256);
    const unsigned gN    = (unsigned)((N + 255) / 256);
    const unsigned gAgg  = (unsigned)(((size_t)E * 32 + 255) / 256);

    scalars_kernel<<<1, HIDDEN, 0, stream>>>(meth, hist, logd, SC);

    hipMemsetAsync(DEG, 0, (size_t)N * sizeof(float), stream);
    degree_kernel<<<gE, 256, 0, stream>>>(dst, DEG, E);
    dinv_kernel<<<gN, 256, 0, stream>>>(DEG, N);
    enorm_kernel<<<gE, 256, 0, stream>>>(src, dst, DEG, ENORM, E);

    // input projection -> H0
    gemm128<64, 1><<<gGemm, 256, 0, stream>>>(x, nullptr, in_w, in_b, in_lg, in_lb,
                                              SC, nullptr, nullptr, H0, N);
    // ACC = H0
    hipMemcpyAsync(ACC, H0, (size_t)NH * sizeof(float), hipMemcpyDeviceToDevice, stream);

    auto ode_f = [&](const float* yin) {
        // layer 0
        gemm128<128, 0><<<gGemm, 256, 0, stream>>>(yin, nullptr, gcn_w, nullptr, nullptr,
                                                   nullptr, nullptr, nullptr, nullptr, XW, N);
        hipMemsetAsync(AGG, 0, (size_t)NH * sizeof(float), stream);
        aggregate_kernel<<<gAgg, 256, 0, stream>>>(XW, src, dst, ENORM, AGG, E);
        bias_ln_kernel<<<gLN, 256, 0, stream>>>(AGG, gcn_b, ln_g, ln_b, CUR, N);
        // layers 1,2 with gating
        for (int i = 1; i < 3; ++i) {
            gemm128<128, 0><<<gGemm, 256, 0, stream>>>(
                CUR, nullptr, gcn_w + (size_t)i * HIDDEN * HIDDEN, nullptr, nullptr,
                nullptr, nullptr, nullptr, nullptr, XW, N);
            hipMemsetAsync(AGG, 0, (size_t)NH * sizeof(float), stream);
            aggregate_kernel<<<gAgg, 256, 0, stream>>>(XW, src, dst, ENORM, AGG, E);
            bias_ln_kernel<<<gLN, 256, 0, stream>>>(AGG, gcn_b + i * HIDDEN,
                                                    ln_g + i * HIDDEN, ln_b + i * HIDDEN,
                                                    AGG, N);
            gemm128<256, 2><<<gGemm, 256, 0, stream>>>(CUR, AGG, gate_w, gate_b, nullptr,
                                                       nullptr, nullptr, nullptr, nullptr,
                                                       CUR, N);
        }
    };

    const float carr[4] = {0.5f, 0.5f, 1.0f, 0.0f};
    const float warr[4] = {1.f, 2.f, 2.f, 1.f};
    const float* yin = H0;
    for (int s = 0; s < 4; ++s) {
        ode_f(yin);  // pre-tanh k_s in CUR
        tanh_rk4_kernel<<<gEl, 256, 0, stream>>>(CUR, yin, H0, Y, ACC, rw, SC,
                                                 carr[s], warr[s], NH);
        yin = Y;
    }

    // output projection + LN + mean pool
    hipMemsetAsync(POOL, 0, (size_t)G * HIDDEN * sizeof(float), stream);
    hipMemsetAsync(CNT, 0, (size_t)G * sizeof(float), stream);
    gemm128<128, 3><<<gGemm, 256, 0, stream>>>(ACC, nullptr, out_w, out_b, o_lg, o_lb,
                                               SC, batchv, POOL, nullptr, N);
    count_kernel<<<gN, 256, 0, stream>>>(batchv, CNT, N);
    finalize_kernel<<<(unsigned)((G * HIDDEN + 255) / 256), 256, 0, stream>>>(
        POOL, CNT, (float*)d_out, G * HIDDEN);
}